// NordModel_13211319403052
// MI455X (gfx1250) — compile-verified
//
#include <hip/hip_runtime.h>
#include <math.h>

// Problem constants (match reference)
#define NCq     64
#define Dq      1024
#define Tq      8
#define Bq      4096
#define ROWS    16          // batch rows per block
#define NTHREADS 1024
#define V_RESETq (-0.1f)
#define REFRAC_Tq 2

#define WSTRIDE 68          // padded LDS stride for W (64x64)
#define CSTRIDE 68          // padded LDS stride for cf / nb (16x64)

typedef float v2f __attribute__((ext_vector_type(2)));
typedef float v8f __attribute__((ext_vector_type(8)));

__launch_bounds__(NTHREADS)
__global__ void snn_cascade_fused(const float* __restrict__ cur,
                                  const float* __restrict__ thr,
                                  const float* __restrict__ bmemraw,
                                  const float* __restrict__ bsynraw,
                                  const float* __restrict__ nw,
                                  const float* __restrict__ gain,
                                  float* __restrict__ out_s,
                                  float* __restrict__ out_v)
{
    __shared__ float Wl[NCq * WSTRIDE];      // W[c][k] = sigmoid(nw[c][k])
    __shared__ float cf[ROWS * CSTRIDE];     // cluster fire counts (un-scaled)
    __shared__ float nb[ROWS * CSTRIDE];     // neighbor (post gain*scale)
    __shared__ float gl[NCq];                // gain[c] * scale

    const int tid  = threadIdx.x;            // == feature index d
    const int b0   = blockIdx.x * ROWS;
    const int cidx = tid & (NCq - 1);        // cluster id of this feature

    const float bm = 1.0f / (1.0f + __expf(-bmemraw[0]));
    const float bs = 1.0f / (1.0f + __expf(-bsynraw[0]));
    const float one_m_bm = 1.0f - bm;
    const float th = thr[tid];

    // Stage W = sigmoid(neighbor_weights) into LDS, row-major [c][k], padded.
    for (int i = tid; i < NCq * NCq; i += NTHREADS) {
        int r = i >> 6, c = i & 63;
        Wl[r * WSTRIDE + c] = 1.0f / (1.0f + __expf(-nw[i]));
    }
    // Fold scale = 1/(D/NC) = 1/16 (exact power of two) into the gain.
    if (tid < NCq) gl[tid] = gain[tid] * (1.0f / 16.0f);

    // Recurrent state lives in registers for the whole kernel.
    float v[ROWS], isyn[ROWS];
    int   refrac[ROWS];
#pragma unroll
    for (int r = 0; r < ROWS; ++r) { v[r] = 0.0f; isyn[r] = 0.0f; refrac[r] = 0; }

    __syncthreads();

    const int lane  = tid & 31;
    const int wid   = tid >> 5;
    const int mrow  = lane & 15;             // M (or N) index within tile
    const int khalf = (lane >> 4) << 1;      // 0 for lanes 0-15, 2 for lanes 16-31
    const int rbase = (lane >> 4) << 3;      // C/D row base: 0 or 8

    for (int t = 0; t < Tq; ++t) {
        // ---- zero cluster-fire accumulators -------------------------------
        cf[tid] = 0.0f;
        if (tid < ROWS * CSTRIDE - NTHREADS) cf[NTHREADS + tid] = 0.0f;
        __syncthreads();

        // ---- phase 1: synapse/membrane update + spike + cluster reduce ----
        const float* curt = cur + ((size_t)t * Bq + b0) * Dq + tid;
        float s[ROWS];
#pragma unroll
        for (int r = 0; r < ROWS; ++r) {
            float c  = curt[(size_t)r * Dq];
            isyn[r]  = bs * isyn[r] + c;
            float vm = bm * v[r] + one_m_bm * isyn[r];
            v[r]     = (refrac[r] > 0) ? V_RESETq : vm;
            s[r]     = (v[r] >= th) ? 1.0f : 0.0f;
            if (s[r] > 0.0f)
                atomicAdd(&cf[r * CSTRIDE + cidx], 1.0f);   // ds_add_f32
        }
        // Prefetch next timestep's contiguous 64KB tile (64B per thread).
        if (t + 1 < Tq) {
            const float* pf = cur + ((size_t)(t + 1) * Bq + b0) * Dq + (size_t)tid * 16;
            __builtin_prefetch(pf, 0, 1);                   // global_prefetch_b8
        }
        __syncthreads();

        // ---- phase 2: neighbor[r][c] = sum_k cf[r][k] * W[c][k], * gain*scale
        // Waves 0..3 each own one 16-wide N tile; f32 WMMA 16x16x4, exact math.
        if (wid < 4) {
            const int c0 = wid << 4;
            v8f acc = {0.f, 0.f, 0.f, 0.f, 0.f, 0.f, 0.f, 0.f};
#pragma unroll
            for (int k0 = 0; k0 < NCq; k0 += 4) {
                v2f a, b;
                // A (16x4): lane<16 -> M=lane, K=k0+{0,1}; lane>=16 -> K=k0+{2,3}
                a.x = cf[mrow * CSTRIDE + k0 + khalf];
                a.y = cf[mrow * CSTRIDE + k0 + khalf + 1];
                // B (4x16): B[k][n] = W[c0+n][k0+k]; N = lane (mod 16)
                b.x = Wl[(c0 + mrow) * WSTRIDE + k0 + khalf];
                b.y = Wl[(c0 + mrow) * WSTRIDE + k0 + khalf + 1];
                acc = __builtin_amdgcn_wmma_f32_16x16x4_f32(
                          false, a, false, b, (short)0, acc, false, false);
            }
            const float gv = gl[c0 + mrow];
            // C/D layout: VGPR i -> row i (lanes 0-15) or i+8 (lanes 16-31), col = lane%16
#pragma unroll
            for (int i = 0; i < 8; ++i)
                nb[(rbase + i) * CSTRIDE + c0 + mrow] = acc[i] * gv;
        }
        __syncthreads();

        // ---- phase 3: cascade into i_syn, reset, refractory, store outputs
        float* os = out_s + ((size_t)t * Bq + b0) * Dq + tid;
        float* ov = out_v + ((size_t)t * Bq + b0) * Dq + tid;
#pragma unroll
        for (int r = 0; r < ROWS; ++r) {
            isyn[r] += nb[r * CSTRIDE + cidx];
            v[r]    -= s[r] * th;
            refrac[r] = (s[r] > 0.0f) ? REFRAC_Tq : (refrac[r] > 0 ? refrac[r] - 1 : 0);
            os[(size_t)r * Dq] = s[r];
            ov[(size_t)r * Dq] = v[r];
        }
        // No trailing barrier needed: the post-WMMA barrier orders the WMMA cf reads before the next
        // cf zeroing; the post-zeroing barrier orders this iteration's nb reads before the next WMMA.
    }
}

extern "C" void kernel_launch(void* const* d_in, const int* in_sizes, int n_in,
                              void* d_out, int out_size, void* d_ws, size_t ws_size,
                              hipStream_t stream) {
    const float* cur  = (const float*)d_in[0];   // (T,B,D)
    const float* thr  = (const float*)d_in[1];   // (D,)
    const float* bmem = (const float*)d_in[2];   // scalar
    const float* bsyn = (const float*)d_in[3];   // scalar
    const float* nw   = (const float*)d_in[4];   // (NC,NC)
    const float* gain = (const float*)d_in[5];   // (NC,)

    float* out   = (float*)d_out;
    float* out_s = out;                                   // spikes (T,B,D)
    float* out_v = out + (size_t)Tq * Bq * Dq;            // v_trace (T,B,D)

    dim3 grid(Bq / ROWS);      // 256 blocks
    dim3 block(NTHREADS);      // 1024 threads = 32 wave32s
    hipLaunchKernelGGL(snn_cascade_fused, grid, block, 0, stream,
                       cur, thr, bmem, bsyn, nw, gain, out_s, out_v);
}